// GAT_57775900066538
// MI455X (gfx1250) — compile-verified
//
#include <hip/hip_runtime.h>
#include <hip/hip_bf16.h>

// ---------------- problem constants ----------------
#define Bc   16
#define Nc   5000
#define Ec   160000
#define INF_ 64
#define OUTF 64
#define BNc  (Bc * Nc)          // 80000
#define BEc  (Bc * Ec)          // 2,560,000
#define ETOT (BEc + BNc)        // 2,640,000
#define OUT_ELEMS (BNc * OUTF)  // 5,120,000

typedef __attribute__((ext_vector_type(2))) float v2f;
typedef __attribute__((ext_vector_type(8))) float v8f;

// ---------------- exact float atomic max via sign-split ----------------
__device__ __forceinline__ void atomicMaxFloat(float* addr, float val) {
    // works given *addr initialized to -inf (0xFF800000)
    if (val >= 0.0f) atomicMax((int*)addr, __float_as_int(val));
    else             atomicMin((unsigned int*)addr, __float_as_uint(val));
}

// ---------------- K0: init out(+bias), amax(-inf), denom(0) ----------------
__global__ __launch_bounds__(256) void k_init(float* __restrict__ out,
                                              const float* __restrict__ bias,
                                              float* __restrict__ amax,
                                              float* __restrict__ denom) {
    int i = blockIdx.x * 256 + threadIdx.x;
    if (i < OUT_ELEMS) out[i] = bias[i & 63];
    if (i < BNc) { amax[i] = -INFINITY; denom[i] = 0.0f; }
}

// ---------------- K0b: s_edge = dot(W_edge, att_edge); ea_mean ----------------
__global__ __launch_bounds__(256) void k_scalars(const float* __restrict__ W_edge,
                                                 const float* __restrict__ att_edge,
                                                 const float* __restrict__ edge_attr,
                                                 float* __restrict__ sc) {
    __shared__ float red[256];
    int t = threadIdx.x;
    float se = (t < 64) ? W_edge[t] * att_edge[t] : 0.0f;
    red[t] = se; __syncthreads();
    for (int s = 128; s > 0; s >>= 1) { if (t < s) red[t] += red[t + s]; __syncthreads(); }
    float s_edge = red[0];
    __syncthreads();
    float acc = 0.0f;
    for (int i = t; i < Ec; i += 256) acc += edge_attr[i];
    red[t] = acc; __syncthreads();
    for (int s = 128; s > 0; s >>= 1) { if (t < s) red[t] += red[t + s]; __syncthreads(); }
    if (t == 0) { sc[0] = s_edge; sc[1] = red[0] * (1.0f / (float)Ec); }
}

// ---------------- K1: xl = x @ W^T via V_WMMA_F32_16X16X4_F32 ----------------
// block = 256 threads = 8 waves; each wave computes one 16x64 output tile.
__global__ __launch_bounds__(256) void k_gemm_wmma(const float* __restrict__ x,
                                                   const float* __restrict__ W,
                                                   float* __restrict__ xl) {
    // B-matrix staged as float2 K-pairs: sB[k2*64 + n] = { W[n][2k2], W[n][2k2+1] }
    __shared__ v2f sB[32 * 64];            // 16 KB
    __shared__ v2f sA[8][16 * 33];         // 8 waves * 528 * 8B = 33 KB (pitch 33 kills conflicts)

    int t = threadIdx.x;
    // cooperative transpose-load of W (64x64) into sB
    #pragma unroll
    for (int i = 0; i < 8; ++i) {
        int idx = i * 256 + t;             // 2048 float2 elements
        int k2 = idx >> 6, n = idx & 63;
        const float* wp = W + n * 64 + k2 * 2;
        v2f val; val.x = wp[0]; val.y = wp[1];
        sB[k2 * 64 + n] = val;
    }

    int wave = t >> 5, lane = t & 31;
    int lo = lane & 15, hi = lane >> 4;
    int rowbase = (blockIdx.x * 8 + wave) * 16;

    // wave loads its 16x64 x-tile (coalesced float4), stores as float2 K-pairs
    const float4* xrow = (const float4*)(x + (long)rowbase * 64);
    #pragma unroll
    for (int i = 0; i < 8; ++i) {
        int flat = i * 32 + lane;          // 0..255 -> (row, c4)
        int r = flat >> 4, c4 = flat & 15;
        float4 v = xrow[r * 16 + c4];
        v2f p0; p0.x = v.x; p0.y = v.y;
        v2f p1; p1.x = v.z; p1.y = v.w;
        sA[wave][r * 33 + c4 * 2 + 0] = p0;
        sA[wave][r * 33 + c4 * 2 + 1] = p1;
    }
    __syncthreads();

    v8f acc[4];
    #pragma unroll
    for (int nt = 0; nt < 4; ++nt)
        #pragma unroll
        for (int i = 0; i < 8; ++i) acc[nt][i] = 0.0f;

    // A frag (16x4 f32): lanes0-15 row=lane, VGPR0=K0,VGPR1=K1; lanes16-31 VGPR0=K2,VGPR1=K3
    // B frag (4x16 f32): lanes0-15 n=lane,  VGPR0=K0,VGPR1=K1; lanes16-31 VGPR0=K2,VGPR1=K3
    #pragma unroll
    for (int ks = 0; ks < 16; ++ks) {
        v2f a = sA[wave][lo * 33 + ks * 2 + hi];       // row=lo, K = 4ks+2hi, +1
        int brow = (ks * 2 + hi) * 64;
        #pragma unroll
        for (int nt = 0; nt < 4; ++nt) {
            v2f b = sB[brow + nt * 16 + lo];           // n = nt*16+lo, K = 4ks+2hi, +1
            acc[nt] = __builtin_amdgcn_wmma_f32_16x16x4_f32(
                false, a, false, b, (short)0, acc[nt], false, false);
        }
    }

    // C/D layout: VGPR v -> rows {v, v+8}, lanes0-15 N=lane
    float* orow = xl + (long)rowbase * 64;
    #pragma unroll
    for (int nt = 0; nt < 4; ++nt)
        #pragma unroll
        for (int v = 0; v < 8; ++v)
            orow[(v + 8 * hi) * 64 + nt * 16 + lo] = acc[nt][v];
}

// ---------------- K2: a_src/a_dst per node (wave per node) ----------------
__global__ __launch_bounds__(256) void k_node_coeffs(const float* __restrict__ xl,
                                                     const float* __restrict__ att_src,
                                                     const float* __restrict__ att_dst,
                                                     float* __restrict__ a_src,
                                                     float* __restrict__ a_dst) {
    int wave = threadIdx.x >> 5, lane = threadIdx.x & 31;
    int v = blockIdx.x * 8 + wave;
    if (v >= BNc) return;
    float2 xv = ((const float2*)(xl + (long)v * 64))[lane];
    float2 as = ((const float2*)att_src)[lane];
    float2 ad = ((const float2*)att_dst)[lane];
    float ps = xv.x * as.x + xv.y * as.y;
    float pd = xv.x * ad.x + xv.y * ad.y;
    #pragma unroll
    for (int off = 16; off > 0; off >>= 1) {
        ps += __shfl_xor(ps, off, 32);
        pd += __shfl_xor(pd, off, 32);
    }
    if (lane == 0) { a_src[v] = ps; a_dst[v] = pd; }
}

// ---------------- K3: alpha_raw + segment_max; emit src/dst ----------------
__global__ __launch_bounds__(256) void k_edge_p1(const int* __restrict__ ei,
                                                 const float* __restrict__ edge_attr,
                                                 const float* __restrict__ a_src,
                                                 const float* __restrict__ a_dst,
                                                 const float* __restrict__ sc,
                                                 float* __restrict__ amax,
                                                 float* __restrict__ alpha,
                                                 int* __restrict__ srcdst) {
    int e = blockIdx.x * 256 + threadIdx.x;
    if (e >= ETOT) return;
    float s_edge = sc[0];
    int s, d; float aed;
    if (e < BEc) {
        int eo = e % Ec;
        s = ei[eo]; d = ei[Ec + eo];
        aed = edge_attr[eo] * s_edge;
    } else {
        s = d = e - BEc;
        aed = sc[1] * s_edge;                 // mean edge_attr on self-loops
    }
    float al = a_src[s] + a_dst[d] + aed;
    al = (al > 0.0f) ? al : 0.2f * al;        // leaky_relu(0.2)
    alpha[e] = al;
    atomicMaxFloat(amax + d, al);
    srcdst[e] = s;
    srcdst[ETOT + e] = d;
}

// ---------------- K4: exp(alpha - amax[dst]) + segment_sum ----------------
__global__ __launch_bounds__(256) void k_edge_p2(const int* __restrict__ srcdst,
                                                 const float* __restrict__ amax,
                                                 float* __restrict__ alpha,
                                                 float* __restrict__ denom) {
    int e = blockIdx.x * 256 + threadIdx.x;
    if (e >= ETOT) return;
    int d = srcdst[ETOT + e];
    float ex = __expf(alpha[e] - amax[d]);
    alpha[e] = ex;
    atomicAdd(denom + d, ex);
}

// ---------------- K5: normalize + scatter messages (wave per edge) ----------------
__global__ __launch_bounds__(256) void k_edge_scatter(const int* __restrict__ srcdst,
                                                      const float* __restrict__ denom,
                                                      const float* __restrict__ xl,
                                                      float* __restrict__ alpha,
                                                      float* __restrict__ out) {
    int wave = threadIdx.x >> 5, lane = threadIdx.x & 31;
    int e = blockIdx.x * 8 + wave;
    if (e >= ETOT) return;
    int s = srcdst[e];
    int d = srcdst[ETOT + e];
    __builtin_prefetch(xl + (long)s * 64, 0, 0);        // global_prefetch_b8
    float al = alpha[e] / (denom[d] + 1e-16f);
    float2 xv = ((const float2*)(xl + (long)s * 64))[lane];
    if (lane == 0) alpha[e] = al;                       // final normalized alpha output
    float* op = out + (long)d * 64 + 2 * lane;
    atomicAdd(op + 0, xv.x * al);
    atomicAdd(op + 1, xv.y * al);
}

// ---------------- launcher ----------------
extern "C" void kernel_launch(void* const* d_in, const int* in_sizes, int n_in,
                              void* d_out, int out_size, void* d_ws, size_t ws_size,
                              hipStream_t stream) {
    const float* data      = (const float*)d_in[0];   // [B,N,64]
    const int*   ei        = (const int*)  d_in[1];   // [2,E]
    const float* edge_attr = (const float*)d_in[2];   // [E]
    const float* W         = (const float*)d_in[3];   // [64,64]
    const float* W_edge    = (const float*)d_in[4];   // [64,1]
    const float* att_src   = (const float*)d_in[5];   // [1,1,64]
    const float* att_dst   = (const float*)d_in[6];   // [1,1,64]
    const float* att_edge  = (const float*)d_in[7];   // [1,1,64]
    const float* bias      = (const float*)d_in[8];   // [64]

    // workspace layout (floats)
    float* ws     = (float*)d_ws;
    float* xl     = ws;                       // 5,120,000
    float* a_src  = ws + 5120000;             // 80,000
    float* a_dst  = ws + 5200000;             // 80,000
    float* amax   = ws + 5280000;             // 80,000
    float* denom  = ws + 5360000;             // 80,000
    float* sc     = ws + 5440000;             // 2 scalars

    // output layout: [out 5.12M f32 | src,dst 5.28M i32 | alpha 2.64M f32]
    float* out    = (float*)d_out;
    int*   srcdst = (int*)d_out + OUT_ELEMS;
    float* alpha  = (float*)d_out + OUT_ELEMS + 2 * ETOT;

    k_init        <<<OUT_ELEMS / 256, 256, 0, stream>>>(out, bias, amax, denom);
    k_scalars     <<<1, 256, 0, stream>>>(W_edge, att_edge, edge_attr, sc);
    k_gemm_wmma   <<<BNc / 128, 256, 0, stream>>>(data, W, xl);
    k_node_coeffs <<<BNc / 8, 256, 0, stream>>>(xl, att_src, att_dst, a_src, a_dst);
    k_edge_p1     <<<(ETOT + 255) / 256, 256, 0, stream>>>(ei, edge_attr, a_src, a_dst,
                                                           sc, amax, alpha, srcdst);
    k_edge_p2     <<<(ETOT + 255) / 256, 256, 0, stream>>>(srcdst, amax, alpha, denom);
    k_edge_scatter<<<ETOT / 8, 256, 0, stream>>>(srcdst, denom, xl, alpha, out);
}